// GCN_14886356648680
// MI455X (gfx1250) — compile-verified
//
#include <hip/hip_runtime.h>

typedef __attribute__((ext_vector_type(2))) float v2f;
typedef __attribute__((ext_vector_type(8))) float v8f;

#define NCOL 256  // HID == OUT == 256, weight matrices are K x 256 row-major

// Select row pointer from the logical concat([A0; A1]) without materializing it.
__device__ __forceinline__ const float* row_ptr(const float* __restrict__ A0,
                                                const float* __restrict__ A1,
                                                int nrows0, int K, int r) {
  return (r < nrows0) ? (A0 + (size_t)r * K) : (A1 + (size_t)(r - nrows0) * K);
}

// C[M x 256] = act(A[M x K]) @ W[K x 256] (+ bias), fp32 WMMA 16x16x4.
// Block = 256 threads = 8 waves. Wave w owns a 32-row x 64-col output tile:
// rows m0 = blk*64 + (w/4)*32, cols n0 = (w%4)*64. Two A fragments share the
// four B fragments per k-step -> 8 WMMAs per 10 vmem loads.
template <bool RELU_A>
__global__ __launch_bounds__(256) void gemm_wmma_f32(
    const float* __restrict__ A0, const float* __restrict__ A1, int nrows0,
    int M, int K, const float* __restrict__ W, const float* __restrict__ bias,
    float* __restrict__ C) {
  const int lane = threadIdx.x & 31;
  const int wave = threadIdx.x >> 5;
  const int l16  = lane & 15;
  const int half = lane >> 4;  // 0 -> K pair {0,1}; 1 -> K pair {2,3}
  const int m0 = blockIdx.x * 64 + (wave >> 2) * 32;
  const int n0 = (wave & 3) * 64;
  if (m0 >= M) return;  // M % 32 == 0, so surviving waves have full tiles

  // A fragment layout (16x4 f32): lanes 0..15 hold row m+l, K={k,k+1};
  // lanes 16..31 hold the same rows, K={k+2,k+3}.
  const float* arow0 = row_ptr(A0, A1, nrows0, K, m0 + l16);
  const float* arow1 = row_ptr(A0, A1, nrows0, K, m0 + 16 + l16);

  v8f acc[2][4];
#pragma unroll
  for (int t = 0; t < 4; ++t) {
    const float bv = bias ? bias[n0 + t * 16 + l16] : 0.0f;
#pragma unroll
    for (int g = 0; g < 8; ++g) {
      acc[0][t][g] = bv;
      acc[1][t][g] = bv;
    }
  }

#pragma unroll 2
  for (int k = 0; k < K; k += 4) {
    float a0x = arow0[k + half * 2 + 0];
    float a0y = arow0[k + half * 2 + 1];
    float a1x = arow1[k + half * 2 + 0];
    float a1y = arow1[k + half * 2 + 1];
    if (RELU_A) {
      a0x = fmaxf(a0x, 0.0f);
      a0y = fmaxf(a0y, 0.0f);
      a1x = fmaxf(a1x, 0.0f);
      a1y = fmaxf(a1y, 0.0f);
    }
    v2f a0, a1;
    a0.x = a0x; a0.y = a0y;
    a1.x = a1x; a1.y = a1y;
    const float* wrow = W + (size_t)(k + half * 2) * NCOL;
#pragma unroll
    for (int t = 0; t < 4; ++t) {
      v2f b;  // B fragment (4x16): lanes hold column n, K pairs split by half
      b.x = wrow[n0 + t * 16 + l16];         // W[k + half*2    ][n]
      b.y = wrow[NCOL + n0 + t * 16 + l16];  // W[k + half*2 + 1][n]
      acc[0][t] = __builtin_amdgcn_wmma_f32_16x16x4_f32(
          false, a0, false, b, (short)0, acc[0][t], false, false);
      acc[1][t] = __builtin_amdgcn_wmma_f32_16x16x4_f32(
          false, a1, false, b, (short)0, acc[1][t], false, false);
    }
  }

  // C/D layout: VGPR g holds (M = g + 8*half, N = l16) within each 16x16 tile.
#pragma unroll
  for (int f = 0; f < 2; ++f) {
    const int mbase = m0 + f * 16 + half * 8;
#pragma unroll
    for (int t = 0; t < 4; ++t) {
      float* crow = C + (size_t)mbase * NCOL + (n0 + t * 16 + l16);
#pragma unroll
      for (int g = 0; g < 8; ++g) crow[(size_t)g * NCOL] = acc[f][t][g];
    }
  }
}

// Y[rows[e], :] += vals[e] * X[cols[e], :]  -- one wave per edge, 256 feats.
// Both X and Y buffers (102.4 MB each) are L2-resident (192 MB global L2).
__global__ __launch_bounds__(256) void spmm_atomic(
    const int* __restrict__ rows, const int* __restrict__ cols,
    const float* __restrict__ vals, int n_edges, const float* __restrict__ X,
    float* __restrict__ Y) {
  const int e = blockIdx.x * 8 + (threadIdx.x >> 5);
  if (e >= n_edges) return;
  const int lane = threadIdx.x & 31;
  const int r = rows[e];
  const int c = cols[e];
  const float v = vals[e];
  const float4* xr = (const float4*)(X + (size_t)c * NCOL);
  float* yr = Y + (size_t)r * NCOL + lane * 4;
  const float4 x0 = xr[lane];
  const float4 x1 = xr[lane + 32];
  atomicAdd(yr + 0, v * x0.x);
  atomicAdd(yr + 1, v * x0.y);
  atomicAdd(yr + 2, v * x0.z);
  atomicAdd(yr + 3, v * x0.w);
  atomicAdd(yr + 128 + 0, v * x1.x);
  atomicAdd(yr + 128 + 1, v * x1.y);
  atomicAdd(yr + 128 + 2, v * x1.z);
  atomicAdd(yr + 128 + 3, v * x1.w);
}

__global__ __launch_bounds__(256) void zero_f32(float* __restrict__ p, long n4) {
  long i = (long)blockIdx.x * blockDim.x + threadIdx.x;
  if (i < n4) ((float4*)p)[i] = make_float4(0.f, 0.f, 0.f, 0.f);
}

extern "C" void kernel_launch(void* const* d_in, const int* in_sizes, int n_in,
                              void* d_out, int out_size, void* d_ws,
                              size_t ws_size, hipStream_t stream) {
  const float* user = (const float*)d_in[0];
  const float* item = (const float*)d_in[1];
  const int* er = (const int*)d_in[2];
  const int* ec = (const int*)d_in[3];
  const float* ev = (const float*)d_in[4];
  const float* W0 = (const float*)d_in[5];
  const float* b0 = (const float*)d_in[6];
  const float* Wr = (const float*)d_in[7];
  const float* W1 = (const float*)d_in[8];
  const float* b1 = (const float*)d_in[9];

  const int IN_DIM = 128;
  const int HID = 256;
  const int n_users = in_sizes[0] / IN_DIM;
  const int n_items = in_sizes[1] / IN_DIM;
  const int N = n_users + n_items;  // 100000, divisible by 32
  const int n_edges = in_sizes[2];

  float* buf0 = (float*)d_ws;            // x1, then x2
  float* buf1 = buf0 + (size_t)N * HID;  // res -> spmm1 accumulator
  float* out = (float*)d_out;

  const dim3 blk(256);
  const int gemm_blocks = (N + 63) / 64;
  const int spmm_blocks = (n_edges + 7) / 8;

  // x1 = concat(user,item) @ W0 + b0
  gemm_wmma_f32<false><<<gemm_blocks, blk, 0, stream>>>(
      user, item, n_users, N, IN_DIM, W0, b0, buf0);
  // buf1 = concat(user,item) @ Wres0   (serves as spmm accumulator init)
  gemm_wmma_f32<false><<<gemm_blocks, blk, 0, stream>>>(
      user, item, n_users, N, IN_DIM, Wr, nullptr, buf1);
  // buf1 += A_sparse @ buf0
  spmm_atomic<<<spmm_blocks, blk, 0, stream>>>(er, ec, ev, n_edges, buf0, buf1);
  // x2 = relu(buf1) @ W1 + b1  -> buf0  (relu fused into A load)
  gemm_wmma_f32<true><<<gemm_blocks, blk, 0, stream>>>(
      buf1, buf1, N, N, HID, W1, b1, buf0);
  // out = A_sparse @ buf0
  const long n_out4 = (long)N * HID / 4;
  zero_f32<<<(unsigned)((n_out4 + 255) / 256), blk, 0, stream>>>(out, n_out4);
  spmm_atomic<<<spmm_blocks, blk, 0, stream>>>(er, ec, ev, n_edges, buf0, out);
}